// SAGE_884763263550
// MI455X (gfx1250) — compile-verified
//
#include <hip/hip_runtime.h>

typedef __attribute__((ext_vector_type(2))) float v2f;
typedef __attribute__((ext_vector_type(8))) float v8f;

#define N_NODES 50000
#define N_EDGES 800000
#define IN_F    128
#define HID_F   256
#define OUT_F   128

// ---------------------------------------------------------------------------
// Zero a float buffer (grid-stride).
// ---------------------------------------------------------------------------
__global__ void sage_zero_kernel(float* __restrict__ p, long long count) {
    long long i = (long long)blockIdx.x * blockDim.x + threadIdx.x;
    long long stride = (long long)gridDim.x * blockDim.x;
    for (; i < count; i += stride) p[i] = 0.0f;
}

// ---------------------------------------------------------------------------
// In-degree accumulation: deg[dst[e]] += 1  (f32 atomics, L2-resident).
// ---------------------------------------------------------------------------
__global__ void sage_deg_kernel(const int* __restrict__ dst, float* __restrict__ deg) {
    int e = blockIdx.x * blockDim.x + threadIdx.x;
    if (e < N_EDGES) atomicAdd(&deg[dst[e]], 1.0f);
}

// deg[n] -> 1 / max(deg[n], 1)  (in place)
__global__ void sage_invdeg_kernel(float* __restrict__ deg) {
    int n = blockIdx.x * blockDim.x + threadIdx.x;
    if (n < N_NODES) deg[n] = 1.0f / fmaxf(deg[n], 1.0f);
}

// ---------------------------------------------------------------------------
// Edge scatter-add: agg[dst[e], :] += feat[src[e], :]
// One block per edge; blockDim == F so each row gather is one coalesced
// 512B/1KB burst; the adds are global_atomic_add_f32 resolving in L2.
// ---------------------------------------------------------------------------
__global__ void sage_scatter_kernel(const float* __restrict__ feat,
                                    const int*   __restrict__ src,
                                    const int*   __restrict__ dst,
                                    float*       __restrict__ agg,
                                    int F) {
    int e = blockIdx.x;
    int f = threadIdx.x;
    int s = src[e];
    int d = dst[e];
    float v = feat[(size_t)s * F + f];
    atomicAdd(&agg[(size_t)d * F + f], v);
}

// ---------------------------------------------------------------------------
// Fused dual-GEMM:  out = act( A_self @ W_self + (A_agg * inv_deg) @ W_neigh + b )
//
// f32 WMMA (v_wmma_f32_16x16x4_f32). Compile-time K/N so all inner-loop
// load offsets are 24-bit immediates. Each wave computes a 16 x (NSUB*16)
// strip with NSUB independent v8f accumulators reusing one A fragment per
// K-step (register blocking: NSUB-deep independence hides WMMA D->A RAW).
// Block = 128 threads = 4 waves covering the full N of one 16-row tile.
// M % 16 == 0 so EXEC is all-1s for every WMMA (ISA requirement).
// ---------------------------------------------------------------------------
template <int K, int N, int NSUB>
__global__ __launch_bounds__(128)
void sage_gemm_kernel(const float* __restrict__ A_self,   // [M x K] row-major
                      const float* __restrict__ A_agg,    // [M x K] unnormalized sums
                      const float* __restrict__ inv_deg,  // [M]
                      const float* __restrict__ W_self,   // [K x N] row-major
                      const float* __restrict__ W_neigh,  // [K x N]
                      const float* __restrict__ bias,     // [N]
                      float*       __restrict__ out,      // [M x N]
                      int do_relu) {
    const int lane = threadIdx.x & 31;
    const int wave = threadIdx.x >> 5;

    const int m0 = blockIdx.x * 16;
    const int n0 = wave * (NSUB * 16);

    // ISA f32 fragment layouts (CDNA5 ISA):
    //  A 16x4 : lanes 0-15 hold K=0 (v0), K=1 (v1); lanes 16-31 hold K=2, K=3
    //  B 4x16 : lanes 0-15 hold rows K=0 (v0), K=1 (v1); lanes 16-31 K=2, K=3
    //  C/D    : v[i]: lanes 0-15 -> (M=i, N=lane); lanes 16-31 -> (M=8+i, N=lane-16)
    const int r  = lane & 15;            // A row / B-C column within tile
    const int kh = (lane >> 4) << 1;     // 0 or 2 : lane-half K offset

    const float scale = inv_deg[m0 + r]; // mean normalization, fused into pass 2

    v8f acc[NSUB];
#pragma unroll
    for (int s = 0; s < NSUB; ++s) acc[s] = (v8f){0.f,0.f,0.f,0.f,0.f,0.f,0.f,0.f};

    // ---- pass 1: A_self @ W_self -------------------------------------
    {
        const float* a_base = A_self + (size_t)(m0 + r) * K + kh;
        const float* b_base = W_self + (size_t)kh * N + n0 + r;
#pragma unroll 8
        for (int k = 0; k < K; k += 4) {
            v2f a;
            a.x = a_base[k];           // contiguous 64-bit load, imm offset
            a.y = a_base[k + 1];
#pragma unroll
            for (int s = 0; s < NSUB; ++s) {
                v2f b;
                b.x = b_base[k * N + s * 16];        // imm offsets (K,N const)
                b.y = b_base[(k + 1) * N + s * 16];
                acc[s] = __builtin_amdgcn_wmma_f32_16x16x4_f32(
                    false, a, false, b, (short)0, acc[s], false, false);
            }
        }
    }

    // ---- pass 2: (A_agg * inv_deg) @ W_neigh, same accumulators ------
    {
        const float* a_base = A_agg + (size_t)(m0 + r) * K + kh;
        const float* b_base = W_neigh + (size_t)kh * N + n0 + r;
#pragma unroll 8
        for (int k = 0; k < K; k += 4) {
            v2f a;
            a.x = a_base[k] * scale;
            a.y = a_base[k + 1] * scale;
#pragma unroll
            for (int s = 0; s < NSUB; ++s) {
                v2f b;
                b.x = b_base[k * N + s * 16];
                b.y = b_base[(k + 1) * N + s * 16];
                acc[s] = __builtin_amdgcn_wmma_f32_16x16x4_f32(
                    false, a, false, b, (short)0, acc[s], false, false);
            }
        }
    }

    // ---- epilogue: bias (+ ReLU), store ------------------------------
    const int rbase = (lane >> 4) * 8;   // M offset for this lane half
#pragma unroll
    for (int s = 0; s < NSUB; ++s) {
        const float bv = bias[n0 + s * 16 + r];
#pragma unroll
        for (int i = 0; i < 8; ++i) {
            float v = acc[s][i] + bv;
            if (do_relu) v = fmaxf(v, 0.0f);
            out[(size_t)(m0 + rbase + i) * N + (n0 + s * 16 + r)] = v;
        }
    }
}

// ---------------------------------------------------------------------------
// Driver
// ---------------------------------------------------------------------------
extern "C" void kernel_launch(void* const* d_in, const int* in_sizes, int n_in,
                              void* d_out, int out_size, void* d_ws, size_t ws_size,
                              hipStream_t stream) {
    const float* x        = (const float*)d_in[0];
    const int*   src      = (const int*)  d_in[1];
    const int*   dst      = (const int*)  d_in[2];
    const float* W_self1  = (const float*)d_in[3];
    const float* W_neigh1 = (const float*)d_in[4];
    const float* b1       = (const float*)d_in[5];
    const float* W_self2  = (const float*)d_in[6];
    const float* W_neigh2 = (const float*)d_in[7];
    const float* b2       = (const float*)d_in[8];
    float*       out      = (float*)d_out;

    // Workspace layout (floats): deg | agg1 | h | agg2
    float* ws   = (float*)d_ws;
    float* deg  = ws;                                        // N
    float* agg1 = deg  + N_NODES;                            // N * IN_F
    float* h    = agg1 + (size_t)N_NODES * IN_F;             // N * HID_F
    float* agg2 = h    + (size_t)N_NODES * HID_F;            // N * HID_F

    // ---- degree + normalization ----
    {
        long long z1 = (long long)N_NODES * (1 + IN_F);      // deg + agg1 contiguous
        sage_zero_kernel<<<4096, 256, 0, stream>>>(deg, z1);
        sage_deg_kernel<<<(N_EDGES + 255) / 256, 256, 0, stream>>>(dst, deg);
        sage_invdeg_kernel<<<(N_NODES + 255) / 256, 256, 0, stream>>>(deg);
    }

    // ---- layer 1: h = relu(x@Ws1 + mean_agg(x)@Wn1 + b1) ----
    sage_scatter_kernel<<<N_EDGES, IN_F, 0, stream>>>(x, src, dst, agg1, IN_F);
    sage_gemm_kernel<IN_F, HID_F, 4><<<N_NODES / 16, 128, 0, stream>>>(
        x, agg1, deg, W_self1, W_neigh1, b1, h, /*relu=*/1);

    // ---- layer 2: out = h@Ws2 + mean_agg(h)@Wn2 + b2 ----
    sage_zero_kernel<<<4096, 256, 0, stream>>>(agg2, (long long)N_NODES * HID_F);
    sage_scatter_kernel<<<N_EDGES, HID_F, 0, stream>>>(h, src, dst, agg2, HID_F);
    sage_gemm_kernel<HID_F, OUT_F, 2><<<N_NODES / 16, 128, 0, stream>>>(
        h, agg2, deg, W_self2, W_neigh2, b2, out, /*relu=*/0);
}